// GATWithClassifier_29308856828065
// MI455X (gfx1250) — compile-verified
//
#include <hip/hip_runtime.h>
#include <math.h>

#define N_NODES 50000
#define N_EDGES 800000
#define E_TOT   (N_EDGES + N_NODES)
#define HDIM    128
#define HEADS   4
#define HID     32
#define NEG_SLOPE 0.2f
#define EPS_A   1e-16f
#define MONO_NEG_INF 0x007FFFFFu   // f2mono(-inf)

typedef __attribute__((ext_vector_type(2))) float v2f;
typedef __attribute__((ext_vector_type(8))) float v8f;

// Monotone float<->uint mapping so float atomic-max can use atomicMax(uint).
__device__ __forceinline__ unsigned f2mono(float f) {
    unsigned u = __float_as_uint(f);
    return (u & 0x80000000u) ? ~u : (u | 0x80000000u);
}
__device__ __forceinline__ float mono2f(unsigned u) {
    return (u & 0x80000000u) ? __uint_as_float(u & 0x7FFFFFFFu)
                             : __uint_as_float(~u);
}

// ---------------------------------------------------------------------------
// C[M,128] = A[M,128] @ W[128,128]   (fp32, exact) via V_WMMA_F32_16X16X4_F32
// Block = 256 threads = 8 waves; W staged in LDS in K-pair-interleaved layout:
//   wlds_pair[p][n] = (W[2p][n], W[2p+1][n])   (float2, p = K/2)
// so each lane's B fragment (W[k][n], W[k+1][n]) is ONE aligned ds_load_b64
// landing in an even VGPR pair -> feeds v_wmma directly, no register shuffles.
// ---------------------------------------------------------------------------
__global__ __launch_bounds__(256)
void gemm128_wmma(const float* __restrict__ A, const float* __restrict__ W,
                  float* __restrict__ C, int M)
{
    __shared__ float2 wlds[(HDIM / 2) * HDIM];   // 64 KB, pair-interleaved
    for (int i = threadIdx.x; i < (HDIM / 2) * HDIM; i += 256) {
        int p = i >> 7;            // K-pair index
        int n = i & (HDIM - 1);    // column
        float2 w;
        w.x = W[(2 * p) * HDIM + n];       // coalesced across threads (n fast)
        w.y = W[(2 * p + 1) * HDIM + n];
        wlds[i] = w;                        // ds_store_b64
    }
    __syncthreads();

    const int lane = threadIdx.x & 31;
    const int wave = threadIdx.x >> 5;
    const int l16  = lane & 15;
    const int kh   = (lane >> 4) << 1;          // 0 or 2 (K sub-offset per half)

    int row  = blockIdx.x * 128 + wave * 16 + l16;
    int rowc = row < M ? row : (M - 1);         // clamp: keep EXEC all-ones
    const float* arow = A + (size_t)rowc * HDIM;

    v8f acc[8] = {};

    for (int k0 = 0; k0 < HDIM; k0 += 4) {
        // A fragment 16x4 f32: lane M=lane%16, VGPR v holds K = k0+kh+v
        v2f af = *(const v2f*)(arow + k0 + kh);             // global_load_b64
        const float2* wk = wlds + ((k0 + kh) >> 1) * HDIM + l16;
        #pragma unroll
        for (int n = 0; n < 8; ++n) {
            // B fragment 4x16 f32: lane N=n*16+lane%16, VGPR v = K = k0+kh+v
            v2f bf = *(const v2f*)(wk + n * 16);            // ds_load_b64
            acc[n] = __builtin_amdgcn_wmma_f32_16x16x4_f32(
                false, af, false, bf, (short)0, acc[n], false, false);
        }
    }

    // C/D layout: lanes 0-15 -> M = i (VGPR i), lanes 16-31 -> M = 8+i; N = lane%16
    int rbase = blockIdx.x * 128 + wave * 16 + ((lane >> 4) << 3);
    #pragma unroll
    for (int n = 0; n < 8; ++n) {
        int col = n * 16 + l16;
        #pragma unroll
        for (int i = 0; i < 8; ++i) {
            int r = rbase + i;
            if (r < M) C[(size_t)r * HDIM + col] = acc[n][i];
        }
    }
}

// ---------------------------------------------------------------------------
// a_src[n,h] = sum_d h[n,h*32+d]*att_src[h,d];  same for a_dst.
// ---------------------------------------------------------------------------
__global__ void attn_coef(const float* __restrict__ Hf,
                          const float* __restrict__ att_s,
                          const float* __restrict__ att_d,
                          float* __restrict__ a_s, float* __restrict__ a_d)
{
    int t = blockIdx.x * blockDim.x + threadIdx.x;
    if (t >= N_NODES * HEADS) return;
    int node = t >> 2, head = t & 3;
    const float* hp = Hf + (size_t)node * HDIM + head * HID;
    const float* as = att_s + head * HID;
    const float* ad = att_d + head * HID;
    float ss = 0.f, dd = 0.f;
    #pragma unroll
    for (int d = 0; d < HID; ++d) { float v = hp[d]; ss += v * as[d]; dd += v * ad[d]; }
    a_s[t] = ss; a_d[t] = dd;
}

// out buffer pre-seeded with bias (bias added after aggregation in reference);
// m <- -inf (mono-mapped), s <- 0.
__global__ void init_layer(float* __restrict__ outb, const float* __restrict__ bias,
                           unsigned* __restrict__ m, float* __restrict__ s)
{
    int i = blockIdx.x * blockDim.x + threadIdx.x;
    if (i < N_NODES * HDIM) outb[i] = bias[i & (HDIM - 1)];
    if (i < N_NODES * HEADS) { m[i] = MONO_NEG_INF; s[i] = 0.f; }
}

// Pass 1: e = leaky_relu(a_src[src]+a_dst[dst]); store; segment max into m[dst].
__global__ void edge_pass1(const int* __restrict__ ei,
                           const float* __restrict__ a_s, const float* __restrict__ a_d,
                           float* __restrict__ ebuf, unsigned* __restrict__ m)
{
    int e = blockIdx.x * blockDim.x + threadIdx.x;
    if (e >= E_TOT) return;
    int src, dst;
    if (e < N_EDGES) { src = ei[e]; dst = ei[N_EDGES + e]; }
    else             { src = dst = e - N_EDGES; }          // self loops (appended)
    #pragma unroll
    for (int h = 0; h < HEADS; ++h) {
        float v = a_s[src * HEADS + h] + a_d[dst * HEADS + h];
        v = v > 0.f ? v : NEG_SLOPE * v;
        ebuf[(size_t)e * HEADS + h] = v;
        atomicMax(&m[dst * HEADS + h], f2mono(v));
    }
}

// Pass 2: e = exp(e - m[dst]); store; segment sum into s[dst].
__global__ void edge_pass2(const int* __restrict__ ei,
                           const unsigned* __restrict__ m,
                           float* __restrict__ ebuf, float* __restrict__ s)
{
    int e = blockIdx.x * blockDim.x + threadIdx.x;
    if (e >= E_TOT) return;
    int dst = (e < N_EDGES) ? ei[N_EDGES + e] : (e - N_EDGES);
    #pragma unroll
    for (int h = 0; h < HEADS; ++h) {
        float v = expf(ebuf[(size_t)e * HEADS + h] - mono2f(m[dst * HEADS + h]));
        ebuf[(size_t)e * HEADS + h] = v;
        atomicAdd(&s[dst * HEADS + h], v);
    }
}

// Pass 3: alpha = e/(s+eps); out[dst] += alpha * h[src].  One wave per edge;
// each lane owns a contiguous float4 of the 128-wide message (head uniform
// per 8-lane group).
__global__ __launch_bounds__(256)
void edge_pass3(const int* __restrict__ ei,
                const float* __restrict__ ebuf, const float* __restrict__ s,
                const float* __restrict__ Hf, float* __restrict__ outb)
{
    int e = blockIdx.x * 8 + (threadIdx.x >> 5);
    if (e >= E_TOT) return;
    int lane = threadIdx.x & 31;
    int src, dst;
    if (e < N_EDGES) { src = ei[e]; dst = ei[N_EDGES + e]; }
    else             { src = dst = e - N_EDGES; }
    int head = lane >> 3;                       // 4 floats/lane => 8 lanes/head
    float alpha = ebuf[(size_t)e * HEADS + head] /
                  (s[dst * HEADS + head] + EPS_A);
    float4 hv = *(const float4*)(Hf + (size_t)src * HDIM + lane * 4);
    float* op = outb + (size_t)dst * HDIM + lane * 4;
    atomicAdd(op + 0, alpha * hv.x);
    atomicAdd(op + 1, alpha * hv.y);
    atomicAdd(op + 2, alpha * hv.z);
    atomicAdd(op + 3, alpha * hv.w);
}

// Classifier: sigmoid(h @ Wc + bc). Wave per node, shuffle reduction.
__global__ __launch_bounds__(256)
void classifier(const float* __restrict__ Hf, const float* __restrict__ Wc,
                const float* __restrict__ bc, float* __restrict__ out)
{
    int n = blockIdx.x * 8 + (threadIdx.x >> 5);
    if (n >= N_NODES) return;
    int lane = threadIdx.x & 31;
    float4 hv = *(const float4*)(Hf + (size_t)n * HDIM + lane * 4);
    float4 wv = *(const float4*)(Wc + lane * 4);
    float dot = hv.x * wv.x + hv.y * wv.y + hv.z * wv.z + hv.w * wv.w;
    #pragma unroll
    for (int off = 16; off > 0; off >>= 1) dot += __shfl_xor(dot, off, 32);
    if (lane == 0) out[n] = 1.f / (1.f + expf(-(dot + bc[0])));
}

extern "C" void kernel_launch(void* const* d_in, const int* in_sizes, int n_in,
                              void* d_out, int out_size, void* d_ws, size_t ws_size,
                              hipStream_t stream)
{
    const float* x   = (const float*)d_in[0];
    const int*   ei  = (const int*)d_in[1];      // [2, N_EDGES] src row then dst row
    const float* W1  = (const float*)d_in[3];
    const float* as1 = (const float*)d_in[4];
    const float* ad1 = (const float*)d_in[5];
    const float* b1  = (const float*)d_in[6];
    const float* W2  = (const float*)d_in[7];
    const float* as2 = (const float*)d_in[8];
    const float* ad2 = (const float*)d_in[9];
    const float* b2  = (const float*)d_in[10];
    const float* Wc  = (const float*)d_in[11];
    const float* bc  = (const float*)d_in[12];
    float* out = (float*)d_out;

    // Workspace layout (floats): ~68 MB total
    float*    bufA = (float*)d_ws;                            // N*128 (GEMM out)
    float*    bufB = bufA + (size_t)N_NODES * HDIM;           // N*128 (agg out)
    float*    a_s  = bufB + (size_t)N_NODES * HDIM;           // N*4
    float*    a_d  = a_s + (size_t)N_NODES * HEADS;           // N*4
    unsigned* m    = (unsigned*)(a_d + (size_t)N_NODES * HEADS); // N*4
    float*    s    = (float*)m + (size_t)N_NODES * HEADS;     // N*4
    float*    ebuf = s + (size_t)N_NODES * HEADS;             // E_TOT*4

    const int gInit = (N_NODES * HDIM + 255) / 256;
    const int gGemm = (N_NODES + 127) / 128;
    const int gAttn = (N_NODES * HEADS + 255) / 256;
    const int gE    = (E_TOT + 255) / 256;
    const int gE8   = (E_TOT + 7) / 8;
    const int gN8   = (N_NODES + 7) / 8;

    // -------- layer 1 --------
    init_layer <<<gInit, 256, 0, stream>>>(bufB, b1, m, s);
    gemm128_wmma<<<gGemm, 256, 0, stream>>>(x, W1, bufA, N_NODES);
    attn_coef  <<<gAttn, 256, 0, stream>>>(bufA, as1, ad1, a_s, a_d);
    edge_pass1 <<<gE,    256, 0, stream>>>(ei, a_s, a_d, ebuf, m);
    edge_pass2 <<<gE,    256, 0, stream>>>(ei, m, ebuf, s);
    edge_pass3 <<<gE8,   256, 0, stream>>>(ei, ebuf, s, bufA, bufB);
    // -------- layer 2 --------
    gemm128_wmma<<<gGemm, 256, 0, stream>>>(bufB, W2, bufA, N_NODES);
    init_layer <<<gInit, 256, 0, stream>>>(bufB, b2, m, s);
    attn_coef  <<<gAttn, 256, 0, stream>>>(bufA, as2, ad2, a_s, a_d);
    edge_pass1 <<<gE,    256, 0, stream>>>(ei, a_s, a_d, ebuf, m);
    edge_pass2 <<<gE,    256, 0, stream>>>(ei, m, ebuf, s);
    edge_pass3 <<<gE8,   256, 0, stream>>>(ei, ebuf, s, bufA, bufB);
    // -------- classifier --------
    classifier <<<gN8,   256, 0, stream>>>(bufB, Wc, bc, out);
}